// GraphSageEncoder_with_weights_1898375545052
// MI455X (gfx1250) — compile-verified
//
#include <hip/hip_runtime.h>

typedef __attribute__((ext_vector_type(2))) float v2f;
typedef __attribute__((ext_vector_type(8))) float v8f;

constexpr int BATCH   = 8192;
constexpr int KN      = 10;        // neighbors per node
constexpr int FEAT    = 128;       // feature dim
constexpr int EMBED   = 128;       // output dim
constexpr int TILE_M  = 16;        // rows per workgroup
constexpr int XCOLS   = 2 * FEAT;  // 256 = concat(self, neigh)
constexpr int XSTRIDE = XCOLS + 4; // 260: stride%64==4 -> conflict-free column reads

__global__ __launch_bounds__(256)
void sage_fused_kernel(const float* __restrict__ emb,
                       const float* __restrict__ neigh_w,
                       const float* __restrict__ Wm,     // [256,128] row-major
                       const float* __restrict__ bias,   // [128]
                       const int*   __restrict__ nodes,  // [8192]
                       const int*   __restrict__ nids,   // [8192,10]
                       float*       __restrict__ out)    // [8192,128]
{
    __shared__ float sX[TILE_M * XSTRIDE];   // 16.6 KB concat tile
    const int tid     = threadIdx.x;
    const int rowBase = blockIdx.x * TILE_M;

    // ---------- Phase 1: gather self + weighted-mean neighbors into LDS ----------
    {
        const int r  = tid >> 4;            // row within tile (0..15)
        const int c  = (tid & 15) << 3;     // 8-float feature chunk
        const int gr = rowBase + r;

        const float* sp = emb + (size_t)nodes[gr] * FEAT + c;
        float4 s0 = *(const float4*)(sp);
        float4 s1 = *(const float4*)(sp + 4);

        float wk[KN];
        float wsum = 0.f;
        #pragma unroll
        for (int k = 0; k < KN; ++k) { wk[k] = neigh_w[gr * KN + k]; wsum += wk[k]; }
        const float winv = 1.0f / wsum;     // once per thread; precision matters here

        float a[8] = {0.f,0.f,0.f,0.f,0.f,0.f,0.f,0.f};
        #pragma unroll
        for (int k = 0; k < KN; ++k) {
            const float w = wk[k] * winv;
            const float* np = emb + (size_t)nids[gr * KN + k] * FEAT + c;
            float4 n0 = *(const float4*)(np);
            float4 n1 = *(const float4*)(np + 4);
            a[0] += w * n0.x; a[1] += w * n0.y; a[2] += w * n0.z; a[3] += w * n0.w;
            a[4] += w * n1.x; a[5] += w * n1.y; a[6] += w * n1.z; a[7] += w * n1.w;
        }

        float* xr = sX + r * XSTRIDE;
        *(float4*)(xr + c)              = s0;
        *(float4*)(xr + c + 4)          = s1;
        *(float4*)(xr + FEAT + c)       = make_float4(a[0], a[1], a[2], a[3]);
        *(float4*)(xr + FEAT + c + 4)   = make_float4(a[4], a[5], a[6], a[7]);
    }
    __syncthreads();

    // ---------- Phase 2: [16x256] @ [256x128] via V_WMMA_F32_16X16X4_F32 ----------
    {
        const int wave = tid >> 5;          // 0..7 -> 16-column slab
        const int lane = tid & 31;
        const int nloc = lane & 15;
        const int hi   = lane >> 4;         // 0: K,K+1   1: K+2,K+3
        const int koff = hi << 1;
        const int ng   = (wave << 4) + nloc; // global output column
        const int m    = nloc;               // A-fragment row (lane = M)

        v8f acc = {0.f,0.f,0.f,0.f,0.f,0.f,0.f,0.f};
        const float* xrow = sX + m * XSTRIDE;

        #pragma unroll 4
        for (int k = 0; k < XCOLS; k += 4) {
            // A: 16x4 f32 tile from LDS (ISA layout: VGPR0=K+koff, VGPR1=K+koff+1)
            v2f afrag = *(const v2f*)(xrow + k + koff);
            // B: 4x16 f32 tile from W (row-major), symmetric K striping
            v2f bfrag;
            bfrag.x = Wm[(size_t)(k + koff)     * EMBED + ng];
            bfrag.y = Wm[(size_t)(k + koff + 1) * EMBED + ng];
            acc = __builtin_amdgcn_wmma_f32_16x16x4_f32(
                /*neg_a=*/false, afrag, /*neg_b=*/false, bfrag,
                /*c_mod=*/(short)0, acc, /*reuse_a=*/false, /*reuse_b=*/false);
        }

        // Epilogue: bias + swish, C/D layout: VGPR v -> M = v + 8*hi, N = lane&15
        // sigmoid via v_rcp_f32 (no IEEE div_scale chain needed for an activation)
        const float bn = bias[ng];
        #pragma unroll
        for (int v = 0; v < 8; ++v) {
            const int row = rowBase + v + (hi << 3);
            float h   = acc[v] + bn;
            float sig = __builtin_amdgcn_rcpf(1.0f + __expf(-h));
            out[(size_t)row * EMBED + ng] = h * sig;
        }
    }
}

extern "C" void kernel_launch(void* const* d_in, const int* in_sizes, int n_in,
                              void* d_out, int out_size, void* d_ws, size_t ws_size,
                              hipStream_t stream) {
    const float* emb     = (const float*)d_in[0];
    const float* neigh_w = (const float*)d_in[1];
    const float* Wm      = (const float*)d_in[2];
    const float* bias    = (const float*)d_in[3];
    const int*   nodes   = (const int*)d_in[4];
    const int*   nids    = (const int*)d_in[5];
    float* out = (float*)d_out;

    dim3 grid(BATCH / TILE_M);   // 512 workgroups
    dim3 block(256);             // 8 wave32s
    sage_fused_kernel<<<grid, block, 0, stream>>>(emb, neigh_w, Wm, bias, nodes, nids, out);
}